// SinkhornLayer_36344013259273
// MI455X (gfx1250) — compile-verified
//
#include <hip/hip_runtime.h>
#include <math.h>

#define SLICES 64
#define NCOL 1024
#define WGS_PER_SLICE 16
#define ROWS_PER_WG 64                 // NCOL / WGS_PER_SLICE
#define TILE_ELEMS (ROWS_PER_WG * NCOL) // 65536 floats = 256 KB
#define ITERS 10
#define INV_EPS 10.0f

typedef unsigned int u32x4 __attribute__((ext_vector_type(4)));
typedef int i32x8 __attribute__((ext_vector_type(8)));
typedef int i32x4 __attribute__((ext_vector_type(4)));

__device__ __forceinline__ float wave_max(float v) {
#pragma unroll
    for (int off = 16; off; off >>= 1) v = fmaxf(v, __shfl_xor(v, off, 32));
    return v;
}
__device__ __forceinline__ float wave_sum(float v) {
#pragma unroll
    for (int off = 16; off; off >>= 1) v += __shfl_xor(v, off, 32);
    return v;
}

// Sync the 16 workgroups of one slice. Real mechanism: monotonic global
// atomic counter (works for any dispatch). Cluster barrier is issued too:
// it is the native CDNA5 path when cluster-dispatched, and an S_NOP per
// ISA when ClusterID==0, so it is always safe.
__device__ __forceinline__ void slice_barrier(unsigned* cnt, unsigned target) {
    __syncthreads();
    __threadfence();                        // release partial stores
    __builtin_amdgcn_s_cluster_barrier();   // s_barrier_signal/wait -3
    if (threadIdx.x == 0) {
        atomicAdd(cnt, 1u);
        while (__hip_atomic_load(cnt, __ATOMIC_ACQUIRE,
                                 __HIP_MEMORY_SCOPE_AGENT) < target) {
            __builtin_amdgcn_s_sleep(2);
        }
    }
    __syncthreads();
    __threadfence();                        // acquire for all lanes
}

__global__ __launch_bounds__(1024, 1)
void sinkhorn_kernel(const float* __restrict__ Mg, float* __restrict__ out,
                     unsigned* __restrict__ counters) {
    extern __shared__ float smem[];
    float* tile = smem;                 // 64 x 1024
    float* vsh  = smem + TILE_ELEMS;    // 1024
    float* ush  = vsh + NCOL;           // 64

    const int slice = blockIdx.x >> 4;
    const int wg    = blockIdx.x & 15;
    const int tid   = threadIdx.x;
    const int lane  = tid & 31;
    const int wave  = tid >> 5;

    const size_t sliceBase = (size_t)slice * (size_t)(NCOL * NCOL);
    const float* gTile     = Mg + sliceBase + (size_t)wg * TILE_ELEMS;
    float*       gOutSlice = out + sliceBase;
    float*       gPart     = gOutSlice;          // partial scratch (overwritten at end)
    unsigned*    cnt       = counters + slice * 16; // 64B stride per slice

    // ---- TDM: async tensor load of this WG's 64x1024 fp32 tile into LDS ----
    if (wave == 0) {
        unsigned ldsAddr = (unsigned)(size_t)(&tile[0]); // low 32 bits = LDS offset
        unsigned long long ga = (unsigned long long)(size_t)gTile;
        u32x4 g0;
        g0[0] = 1u;                                        // count=1, user D#
        g0[1] = ldsAddr;                                   // lds_addr [63:32]
        g0[2] = (unsigned)(ga & 0xFFFFFFFFull);            // global_addr lo
        g0[3] = (unsigned)((ga >> 32) & 0x01FFFFFFull)     // global_addr [56:32]
              | (2u << 30);                                // type = 2 ("image")
        i32x8 g1;
        g1[0] = (int)(2u << 16);                           // data_size = 4 bytes
        g1[1] = (int)((unsigned)(NCOL & 0xFFFF) << 16);    // tensor_dim0 lo16
        g1[2] = (int)((unsigned)(NCOL >> 16)
              | ((unsigned)(ROWS_PER_WG & 0xFFFF) << 16)); // dim0 hi | tensor_dim1 lo
        g1[3] = (int)((unsigned)(ROWS_PER_WG >> 16)
              | ((unsigned)NCOL << 16));                   // dim1 hi | tile_dim0
        g1[4] = ROWS_PER_WG;                               // tile_dim1 (tile_dim2=0)
        g1[5] = NCOL;                                      // tensor_dim0_stride lo
        g1[6] = 0;
        g1[7] = 0;
        i32x4 z4 = {0, 0, 0, 0};
        i32x8 z8 = {0, 0, 0, 0, 0, 0, 0, 0};
#if __has_builtin(__builtin_amdgcn_tensor_load_to_lds)
        // 6-arg form (clang-23 / therock-10.0): groups 0..3 + extra group + cpol
        __builtin_amdgcn_tensor_load_to_lds(g0, g1, z4, z4, z8, 0);
        __builtin_amdgcn_s_wait_tensorcnt(0);
#endif
    }
    __syncthreads();
#if !__has_builtin(__builtin_amdgcn_tensor_load_to_lds)
    for (int i = tid; i < TILE_ELEMS; i += 1024) tile[i] = gTile[i];
    __syncthreads();
#endif

    // clip & scale in LDS (TDM moved raw data); init v = 0
    for (int i = tid; i < TILE_ELEMS; i += 1024) {
        float x = tile[i];
        tile[i] = fminf(fmaxf(x, -25.0f), 25.0f) * INV_EPS;
    }
    vsh[tid] = 0.0f;
    __syncthreads();

    for (int it = 0; it < ITERS; ++it) {
        // ---- row pass: u[m] = lse_n(tile[m][n] - v[n]); wave w owns rows 2w,2w+1
#pragma unroll
        for (int r = 0; r < 2; ++r) {
            const int m = wave * 2 + r;
            const float* row = tile + m * NCOL;
            float mx = -__builtin_inff();
            for (int c = lane; c < NCOL; c += 32)
                mx = fmaxf(mx, row[c] - vsh[c]);
            mx = wave_max(mx);
            float s = 0.0f;
            for (int c = lane; c < NCOL; c += 32)
                s += __expf(row[c] - vsh[c] - mx);
            s = wave_sum(s);
            if (lane == 0) ush[m] = mx + __logf(s);
        }
        __syncthreads();

        // ---- column partials over this WG's 64 rows; thread owns column n=tid
        {
            const int n = tid;
            float mx = -__builtin_inff();
            for (int m = 0; m < ROWS_PER_WG; ++m)
                mx = fmaxf(mx, tile[m * NCOL + n] - ush[m]);
            float s = 0.0f;
            for (int m = 0; m < ROWS_PER_WG; ++m)
                s += __expf(tile[m * NCOL + n] - ush[m] - mx);
            gPart[wg * 2048 + n]        = mx;
            gPart[wg * 2048 + 1024 + n] = s;
        }

        slice_barrier(cnt, (unsigned)(16 * (2 * it + 1)));

        // ---- combine 16 partials -> v[n] (redundant per WG, identical result)
        {
            const int n = tid;
            float pm[WGS_PER_SLICE], ps[WGS_PER_SLICE];
            float mbig = -__builtin_inff();
#pragma unroll
            for (int w = 0; w < WGS_PER_SLICE; ++w) {
                pm[w] = gPart[w * 2048 + n];
                ps[w] = gPart[w * 2048 + 1024 + n];
                mbig = fmaxf(mbig, pm[w]);
            }
            float S = 0.0f;
#pragma unroll
            for (int w = 0; w < WGS_PER_SLICE; ++w)
                S += ps[w] * __expf(pm[w] - mbig);
            vsh[n] = mbig + __logf(S);
        }

        // guards partial-region reuse next iter AND the final out writes
        slice_barrier(cnt, (unsigned)(16 * (2 * it + 2)));
    }

    // ---- final: out = exp(P0 - u - v), coalesced across columns ----
    {
        const int n = tid;
        const float vn = vsh[n];
        for (int m = 0; m < ROWS_PER_WG; ++m) {
            float x = tile[m * NCOL + n] - ush[m] - vn;
            gOutSlice[(size_t)(wg * ROWS_PER_WG + m) * NCOL + n] = __expf(x);
        }
    }
}

extern "C" void kernel_launch(void* const* d_in, const int* in_sizes, int n_in,
                              void* d_out, int out_size, void* d_ws, size_t ws_size,
                              hipStream_t stream) {
    (void)in_sizes; (void)n_in; (void)out_size; (void)ws_size;
    const float* Mg = (const float*)d_in[0];
    float* out      = (float*)d_out;
    unsigned* cnt   = (unsigned*)d_ws;

    const size_t shmem = (size_t)(TILE_ELEMS + NCOL + 64) * sizeof(float); // ~260 KB
    (void)hipFuncSetAttribute((const void*)sinkhorn_kernel,
                              hipFuncAttributeMaxDynamicSharedMemorySize, (int)shmem);

    // zero the per-slice barrier counters (graph-capture-safe async memset)
    (void)hipMemsetAsync(d_ws, 0, SLICES * 16 * sizeof(unsigned), stream);

    dim3 grid(SLICES * WGS_PER_SLICE);
    dim3 block(1024);
    sinkhorn_kernel<<<grid, block, shmem, stream>>>(Mg, out, cnt);
}